// LQR_12910671692089
// MI455X (gfx1250) — compile-verified
//
#include <hip/hip_runtime.h>
#include <hip/hip_bf16.h>

#define NB  512
#define T   128
#define NS  32
#define NC  16
#define NSC 48

typedef __attribute__((ext_vector_type(2))) float v2f;
typedef __attribute__((ext_vector_type(8))) float v8f;
typedef __attribute__((ext_vector_type(4))) unsigned int u32x4;
typedef __attribute__((ext_vector_type(4))) int i32x4;
typedef __attribute__((ext_vector_type(8))) int i32x8;

// ---- Tensor Data Mover: stage one contiguous 48x48 f32 tile global->LDS ----
// D# per CDNA5 ISA: group0 = {count, lds_addr,
// global_addr[56:0], type=2}; group1 = {data_size=4B, 1-D tile of NSC*NSC
// elements, tensor_dim0 = tile_dim0 = stride = NSC*NSC}.  Tracked by TENSORcnt.
__device__ __forceinline__ void tdm_load_tile(float* dstLds, const float* srcGlobal) {
  unsigned long long ga = (unsigned long long)(const void*)srcGlobal;
  unsigned lds = (unsigned)(unsigned long long)(void*)dstLds;  // low 32b = LDS offset
  u32x4 g0;
  g0[0] = 1u;                                   // count=1 (valid), user mode
  g0[1] = lds;                                  // lds_addr (bytes)
  g0[2] = (unsigned)ga;                         // global_addr[31:0]
  g0[3] = (unsigned)(ga >> 32) | (2u << 30);    // global_addr[56:32] | type=2
  i32x8 g1;
  g1[0] = (2 << 16);                            // data_size = 2 -> 4 bytes
  g1[1] = (int)((unsigned)(NSC * NSC) << 16);   // tensor_dim0[15:0]
  g1[2] = (1 << 16);                            // tensor_dim1 = 1
  g1[3] = (int)((unsigned)(NSC * NSC) << 16);   // tile_dim0
  g1[4] = 0;                                    // tile_dim1/2 unused
  g1[5] = (NSC * NSC);                          // tensor_dim0_stride[31:0]
  g1[6] = 0;
  g1[7] = 0;
  i32x4 z4 = {0, 0, 0, 0};                      // groups 2/3 unused (<=2D)
#if defined(__clang_major__) && (__clang_major__ >= 23)
  i32x8 z8 = {0, 0, 0, 0, 0, 0, 0, 0};
  __builtin_amdgcn_tensor_load_to_lds(g0, g1, z4, z4, z8, 0);
#else
  __builtin_amdgcn_tensor_load_to_lds(g0, g1, z4, z4, 0);
#endif
}

// ---- WMMA fragment loaders (f32 16x16x4, wave32; ISA 7.12.2 layouts) ----
__device__ __forceinline__ v2f ldA(const float* X, int ld, int mr, int kc, int m, int h) {
  const float* r = X + (mr + m) * ld + kc + 2 * h;
  v2f a; a.x = r[0]; a.y = r[1]; return a;
}
__device__ __forceinline__ v2f ldAT(const float* X, int ld, int mr, int kc, int m, int h) {
  v2f a;
  a.x = X[(kc + 2 * h + 0) * ld + mr + m];
  a.y = X[(kc + 2 * h + 1) * ld + mr + m];
  return a;
}
__device__ __forceinline__ v2f ldB(const float* X, int ld, int kc, int nc, int n, int h) {
  v2f b;
  b.x = X[(kc + 2 * h + 0) * ld + nc + n];
  b.y = X[(kc + 2 * h + 1) * ld + nc + n];
  return b;
}
__device__ __forceinline__ v8f ldC(const float* X, int ld, int mr, int nc, int n, int h) {
  v8f c;
#pragma unroll
  for (int v = 0; v < 8; ++v) c[v] = X[(mr + v + 8 * h) * ld + nc + n];
  return c;
}
__device__ __forceinline__ void stC(float* X, int ld, int mr, int nc, int n, int h, v8f c) {
#pragma unroll
  for (int v = 0; v < 8; ++v) X[(mr + v + 8 * h) * ld + nc + n] = c[v];
}

template <bool AT, int K>
__device__ __forceinline__ v8f mmtile(const float* A, int lda, const float* B, int ldb,
                                      int mr, int nc, v8f acc, int lane) {
  const int m = lane & 15, h = lane >> 4;
#pragma unroll
  for (int kk = 0; kk < K; kk += 4) {
    v2f a = AT ? ldAT(A, lda, mr, kk, m, h) : ldA(A, lda, mr, kk, m, h);
    v2f b = ldB(B, ldb, kk, nc, m, h);
    acc = __builtin_amdgcn_wmma_f32_16x16x4_f32(false, a, false, b, (short)0, acc,
                                                false, false);
  }
  return acc;
}

__global__ __launch_bounds__(128) void lqr_kernel(
    const float* __restrict__ xinit, const float* __restrict__ Q,
    const float* __restrict__ P, const float* __restrict__ Ain,
    const float* __restrict__ Bin, const float* __restrict__ c1in,
    float* __restrict__ out, float* __restrict__ ws) {
  __shared__ __align__(16) float sQbuf[2][NSC * NSC];  // double-buffered Qt tiles
  __shared__ float sF[NS * NSC];                       // F = [A | Bm]
  __shared__ float sV[NS * NS];                        // V
  __shared__ float sW[NS * NSC];                       // W = V@F
  __shared__ float sq[NSC];
  __shared__ float sv[NS];
  __shared__ float sz[NS];
  __shared__ float sc1[NS];
  __shared__ float skt[NC];
  __shared__ float sfac[NC];
  __shared__ float sred[NSC];
  __shared__ float sx[NSC];
  __shared__ float spiv;

  const int b = blockIdx.x;
  const int tid = threadIdx.x;
  const int lane = tid & 31;
  const int wave = tid >> 5;
  const int m16 = lane & 15;
  const int hh = lane >> 4;
  const float* Qbase = Q + (size_t)b * T * (NSC * NSC);

  // ---- init: F, c1, V=0, v=0 ----
  for (int e = tid; e < NS * NS; e += 128) {
    sF[(e >> 5) * NSC + (e & 31)] = Ain[(size_t)b * NS * NS + e];
    sV[e] = 0.0f;
  }
  for (int e = tid; e < NS * NC; e += 128)
    sF[(e >> 4) * NSC + NS + (e & 15)] = Bin[(size_t)b * NS * NC + e];
  if (tid < NS) { sc1[tid] = c1in[(size_t)b * NS + tid]; sv[tid] = 0.0f; }

  // kick off TDM load of the last timestep's Qt (wave-uniform guard: the
  // tensor op is a wave-level instruction and ignores EXEC)
  if (wave == 0) tdm_load_tile(sQbuf[(T - 1) & 1], Qbase + (size_t)(T - 1) * (NSC * NSC));
  __syncthreads();

  // ================= backward Riccati sweep =================
  for (int t = T - 1; t >= 0; --t) {
    float* sQ = sQbuf[t & 1];
    // overlap: DMA next tile into the other buffer while we compute
    if (wave == 0 && t > 0)
      tdm_load_tile(sQbuf[(t - 1) & 1], Qbase + (size_t)(t - 1) * (NSC * NSC));
    // TENSORcnt completes in-order: <=1 outstanding means load(t) is done
    if (t > 0) __builtin_amdgcn_s_wait_tensorcnt(1);
    else       __builtin_amdgcn_s_wait_tensorcnt(0);
    if (tid < NSC) sq[tid] = P[((size_t)b * T + t) * NSC + tid];
    __syncthreads();

    // W = V @ F  (6 tiles of 16x16, K=32)
    for (int ti = wave; ti < 6; ti += 4) {
      int mr = (ti / 3) * 16, nc = (ti % 3) * 16;
      v8f acc = {};
      acc = mmtile<false, NS>(sV, NS, sF, NSC, mr, nc, acc, lane);
      stC(sW, NSC, mr, nc, m16, hh, acc);
    }
    // z = v + V @ c1
    if (tid < NS) {
      float acc = sv[tid];
      for (int k = 0; k < NS; ++k) acc += sV[tid * NS + k] * sc1[k];
      sz[tid] = acc;
    }
    __syncthreads();

    // Qt += F^T @ W  (9 tiles, K=32, C read-modify-write in LDS)
    for (int ti = wave; ti < 9; ti += 4) {
      int mr = (ti / 3) * 16, nc = (ti % 3) * 16;
      v8f acc = ldC(sQ, NSC, mr, nc, m16, hh);
      acc = mmtile<true, NS>(sF, NSC, sW, NSC, mr, nc, acc, lane);
      stC(sQ, NSC, mr, nc, m16, hh, acc);
    }
    // qt = pt + F^T @ z
    if (tid < NSC) {
      float acc = sq[tid];
      for (int i = 0; i < NS; ++i) acc += sF[i * NSC + tid] * sz[i];
      sq[tid] = acc;
    }
    __syncthreads();

    // Gauss-Jordan: Quu * [X | x] = [Qux | qu]  (Quu PD -> no pivoting)
    for (int k = 0; k < NC; ++k) {
      if (tid == 0) spiv = 1.0f / sQ[(NS + k) * NSC + NS + k];
      if (tid < NC && tid != k) sfac[tid] = sQ[(NS + tid) * NSC + NS + k];
      __syncthreads();
      if (tid < NSC) sQ[(NS + k) * NSC + tid] *= spiv;
      else if (tid == NSC) sq[NS + k] *= spiv;
      __syncthreads();
      for (int e = tid; e < NC * (NSC + 1); e += 128) {
        int i = e / (NSC + 1), j = e - i * (NSC + 1);
        if (i == k) continue;
        float f = sfac[i];
        if (j < NSC) sQ[(NS + i) * NSC + j] -= f * sQ[(NS + k) * NSC + j];
        else         sq[NS + i]            -= f * sq[NS + k];
      }
      __syncthreads();
    }

    // Kt = -(Quu^-1 Qux) in-place; kt = -(Quu^-1 qu)
    for (int e = tid; e < NC * NS; e += 128) {
      int i = e >> 5, j = e & 31;
      sQ[(NS + i) * NSC + j] = -sQ[(NS + i) * NSC + j];
    }
    if (tid < NC) skt[tid] = -sq[NS + tid];
    __syncthreads();

    // spill Kt (16x32) + kt (16) for the forward pass
    float* wsb = ws + ((size_t)b * T + t) * 528;
    for (int e = tid; e < NC * NS; e += 128)
      wsb[e] = sQ[(NS + (e >> 5)) * NSC + (e & 31)];
    if (tid < NC) wsb[512 + tid] = skt[tid];

    // v_new = qx + Qxu @ kt
    float vnew = 0.0f;
    if (tid < NS) {
      vnew = sq[tid];
      for (int k = 0; k < NC; ++k) vnew += sQ[tid * NSC + NS + k] * skt[k];
    }
    // V_new = Qxx + Qxu @ Kt  (4 tiles, K=16; one per wave)
    {
      int mr = (wave >> 1) * 16, nc = (wave & 1) * 16;
      v8f acc = ldC(sQ, NSC, mr, nc, m16, hh);
      acc = mmtile<false, NC>(sQ + NS, NSC, sQ + NS * NSC, NSC, mr, nc, acc, lane);
      stC(sV, NS, mr, nc, m16, hh, acc);
    }
    if (tid < NS) sv[tid] = vnew;
    __syncthreads();
  }

  // ================= forward rollout =================
  float* outx = out;
  float* outu = out + (size_t)NB * T * NS;
  float* outc = out + (size_t)NB * T * (NS + NC);
  if (tid < NS) sx[tid] = xinit[(size_t)b * NS + tid];
  float cost = 0.0f;
  if (wave == 0) tdm_load_tile(sQbuf[0], Qbase);  // Qt for t=0
  __syncthreads();

  for (int t = 0; t < T; ++t) {
    const float* sQt = sQbuf[t & 1];
    if (wave == 0 && t + 1 < T)
      tdm_load_tile(sQbuf[(t + 1) & 1], Qbase + (size_t)(t + 1) * (NSC * NSC));
    if (t + 1 < T) __builtin_amdgcn_s_wait_tensorcnt(1);
    else           __builtin_amdgcn_s_wait_tensorcnt(0);

    const float* wsb = ws + ((size_t)b * T + t) * 528;
    // u = Kt @ x + kt
    if (tid < NC) {
      float acc = wsb[512 + tid];
      for (int j = 0; j < NS; ++j) acc += wsb[tid * NS + j] * sx[j];
      sx[NS + tid] = acc;
    }
    __syncthreads();
    // emit x_t, u_t
    if (tid < NS)       outx[((size_t)b * T + t) * NS + tid] = sx[tid];
    else if (tid < NSC) outu[((size_t)b * T + t) * NC + (tid - NS)] = sx[tid];
    // x_next = F @ [x;u] + c1
    float xn = 0.0f;
    if (tid < NS) {
      xn = sc1[tid];
      for (int j = 0; j < NSC; ++j) xn += sF[tid * NSC + j] * sx[j];
    }
    // cost partials from the LDS-resident Qt tile
    const float* prow = P + ((size_t)b * T + t) * NSC;
    if (tid < NSC) {
      float y = 0.0f;
      for (int j = 0; j < NSC; ++j) y += sQt[tid * NSC + j] * sx[j];
      sred[tid] = sx[tid] * (0.5f * y + prow[tid]);
    }
    __syncthreads();
    if (tid == 0) {
      float c = 0.0f;
      for (int j = 0; j < NSC; ++j) c += sred[j];
      cost += c;
    }
    if (tid < NS) sx[tid] = xn;
    __syncthreads();
  }
  if (tid == 0) outc[b] = cost;
}

extern "C" void kernel_launch(void* const* d_in, const int* in_sizes, int n_in,
                              void* d_out, int out_size, void* d_ws, size_t ws_size,
                              hipStream_t stream) {
  const float* xinit = (const float*)d_in[0];
  const float* Q     = (const float*)d_in[1];
  const float* P     = (const float*)d_in[2];
  const float* A     = (const float*)d_in[3];
  const float* Bm    = (const float*)d_in[4];
  const float* c1    = (const float*)d_in[5];
  (void)in_sizes; (void)n_in; (void)out_size; (void)ws_size;
  lqr_kernel<<<NB, 128, 0, stream>>>(xinit, Q, P, A, Bm, c1,
                                     (float*)d_out, (float*)d_ws);
}